// MySDELayer_53618371723906
// MI455X (gfx1250) — compile-verified
//
#include <hip/hip_runtime.h>

#define NTRAJ   512
#define NSTEPS  255
#define TCH     5            // steps per TDM chunk
#define NCH     51           // 51 * 5 = 255
#define ROWF    510          // floats per trajectory row (255 steps * 2 channels)
#define TILEF   (TCH * 2)    // 10 floats per row per chunk
#define BUFB    (32 * TILEF * 4)   // 1280 bytes per wave buffer
#define PARTF   10240        // float index of partials region (16 waves * 2 bufs * 1280B / 4)
#define LDSB    (PARTF * 4 + 48 * 256 * 4)   // 40960 + 49152 = 90112 bytes

typedef unsigned int v4u __attribute__((ext_vector_type(4)));
typedef int          v8i __attribute__((ext_vector_type(8)));
typedef int          v4i __attribute__((ext_vector_type(4)));
typedef float        v2f __attribute__((ext_vector_type(2)));
typedef float        v8f __attribute__((ext_vector_type(8)));

// Issue one TDM 2-D tile load: 32 rows x TILEF floats, row stride ROWF floats,
// from global gptr into LDS byte offset lds_byte_off (contiguous tile rows).
__device__ __forceinline__ void tdm_issue_tile(const float* gptr, unsigned lds_byte_off) {
    unsigned long long ga = (unsigned long long)gptr;
    unsigned ga_lo = __builtin_amdgcn_readfirstlane((unsigned)ga);
    unsigned ga_hi = __builtin_amdgcn_readfirstlane((unsigned)(ga >> 32));
    unsigned lds   = __builtin_amdgcn_readfirstlane(lds_byte_off);
    // D# group0: count=1 | lds_addr | global_addr[56:0] | type=2 (bits 127:126)
    v4u g0 = { 1u, lds, ga_lo, (ga_hi & 0x01FFFFFFu) | (2u << 30) };
    // D# group1: data_size=2(4B); tensor_dim0=510 @bit48; tensor_dim1=512 @bit80;
    // tile_dim0=TILEF @bit112; tile_dim1=32 @bit128; tensor_dim0_stride=510 @bit160;
    // tensor_dim1_stride=510 @bit208 (unused for 2D).
    v8i g1 = { (int)(2u << 16),
               (int)(510u << 16),
               (int)(512u << 16),
               (int)((unsigned)TILEF << 16),
               32,
               510,
               (int)(510u << 16),
               0 };
    v4i gz4 = { 0, 0, 0, 0 };
    v8i gz8 = { 0, 0, 0, 0, 0, 0, 0, 0 };
    __builtin_amdgcn_tensor_load_to_lds(g0, g1, gz4, gz4, gz8, 0);
}

__global__ __launch_bounds__(512) void sde_probs_kernel(const float* __restrict__ inp,
                                                        const float* __restrict__ wv,
                                                        float* __restrict__ out) {
    extern __shared__ float smem[];
    const int tid  = threadIdx.x;
    const int blk  = blockIdx.x;
    const int wave = tid >> 5;
    const int lane = tid & 31;

    // eps = inputs[n % 32] with n = blk*512 + tid, 512 % 32 == 0 -> lane-indexed
    const float eps = inp[lane];

    // Row t = 0 is the deterministic initial state: x=y=0, z=1.
    if (tid < 6) {
        float v = (tid < 4) ? 0.5f : ((tid == 4) ? 1.0f : 0.0f);
        out[(size_t)blk * (256 * 6) + tid] = v;
    }

    const float* grow = wv + ((size_t)blk * NTRAJ + (size_t)wave * 32) * ROWF;
    const unsigned buf0 = (unsigned)(wave * 2) * BUFB;

    float x = 0.0f, y = 0.0f, z = 1.0f;
    const float DTf = 0.00390625f;   // 2^-8
    const float CW  = 0.0375f;       // sqrt(GAMMA) * sqrt(DT) = 0.6 * 0.0625
    const float OM  = 5.0265f;
    const float GA  = 0.36f;

    tdm_issue_tile(grow, buf0);      // prime double buffer
    int s = 0;
    for (int ch = 0; ch < NCH; ++ch) {
        const int cur = ch & 1;
        asm volatile("s_wait_dscnt 0" ::: "memory");   // buffer reuse WAR safety
        if (ch + 1 < NCH) {
            tdm_issue_tile(grow + (size_t)(ch + 1) * TILEF, buf0 + (unsigned)(cur ^ 1) * BUFB);
            __builtin_amdgcn_s_wait_tensorcnt(1);      // oldest TDM (chunk ch) done
        } else {
            __builtin_amdgcn_s_wait_tensorcnt(0);
        }
        const int rb = (wave * 2 + cur) * (BUFB / 4) + lane * TILEF;
#pragma unroll
        for (int si = 0; si < TCH; ++si, ++s) {
            float w0 = smem[rb + 2 * si];
            float w1 = smem[rb + 2 * si + 1];
            float g0 = CW * w0;
            float g1 = CW * w1;
            float xn = x + DTf * (-eps * y - GA * x) - (x * z) * g0 - y * g1;
            float yn = y + DTf * ( eps * x - OM * z - GA * y) - (y * z) * g0 + x * g1;
            float zn = z + (DTf * OM) * y + (1.0f - z * z) * g0;
            x = xn; y = yn; z = zn;
            // wave32 tree reduction of (x, y, z)
            float sx = x, sy = y, sz = z;
            for (int off = 16; off > 0; off >>= 1) {
                sx += __shfl_xor(sx, off, 32);
                sy += __shfl_xor(sy, off, 32);
                sz += __shfl_xor(sz, off, 32);
            }
            if (lane == 0) {
                int pb = PARTF + 48 * s + 3 * wave;   // P[s][wave][c]
                smem[pb]     = sx;
                smem[pb + 1] = sy;
                smem[pb + 2] = sz;
            }
        }
    }
    __syncthreads();

    // Cross-wave reduction as a matmul: O[r] = sum_w P[r][w] = P * ones,
    // rows r = s*3 + c (765 valid rows, padded to 768), K = 16 waves.
    // Exact f32 via chained V_WMMA_F32_16X16X4_F32 (4 K-chunks of 4).
    // Constant trip count (3 tiles/wave) -> no exec-masked loop around WMMA.
    const v2f bone = { 1.0f, 1.0f };
#pragma unroll
    for (int it = 0; it < 3; ++it) {
        const int tile = wave + (it << 4);
        const int r0   = tile << 4;
        const int m    = lane & 15;
        const int r    = r0 + m;           // rows >= 765 are dead padding
        const int sI   = r / 3;
        const int cI   = r - 3 * sI;
        const int kb   = (lane >> 4) << 1; // lanes 0-15: K 0,1 ; lanes 16-31: K 2,3
        v8f acc = {};
#pragma unroll
        for (int k0 = 0; k0 < 16; k0 += 4) {
            v2f a;
            int pb = PARTF + 48 * sI + 3 * (k0 + kb) + cI;
            a.x = smem[pb];
            a.y = smem[pb + 3];
            acc = __builtin_amdgcn_wmma_f32_16x16x4_f32(false, a, false, bone,
                                                        (short)0, acc, false, false);
        }
        // Column 0 of D: lane 0 holds rows r0+0..7, lane 16 holds rows r0+8..15.
        if ((lane & 15) == 0) {
            const int mb = (lane >> 4) * 8;
#pragma unroll
            for (int j = 0; j < 8; ++j) {
                int rr = r0 + mb + j;
                if (rr < 765) {
                    float mean = acc[j] * (1.0f / 512.0f);
                    int t = rr / 3 + 1;
                    int c = rr - 3 * (rr / 3);
                    float p = 0.5f * (mean + 1.0f);
                    size_t o = ((size_t)blk * 256 + (size_t)t) * 6 + (size_t)(2 * c);
                    out[o]     = p;
                    out[o + 1] = 1.0f - p;
                }
            }
        }
    }
}

extern "C" void kernel_launch(void* const* d_in, const int* in_sizes, int n_in,
                              void* d_out, int out_size, void* d_ws, size_t ws_size,
                              hipStream_t stream) {
    (void)in_sizes; (void)n_in; (void)out_size; (void)d_ws; (void)ws_size;
    const float* inp = (const float*)d_in[0];   // (32,1) float32
    const float* wv  = (const float*)d_in[1];   // (16384, 255, 2) float32
    float* out = (float*)d_out;                 // (32, 256, 6) float32
    dim3 grid(32), block(512);
    sde_probs_kernel<<<grid, block, LDSB, stream>>>(inp, wv, out);
}